// DataEmbedding_26886495273553
// MI455X (gfx1250) — compile-verified
//
#include <hip/hip_runtime.h>

typedef __attribute__((ext_vector_type(2))) float v2f;
typedef __attribute__((ext_vector_type(8))) float v8f;

namespace {
constexpr int kSeq    = 4096;
constexpr int kCin    = 7;
constexpr int kNumK   = 74;          // 74 kernels; center-tap weights only
constexpr int kRows   = kSeq * 3;    // 12288 stacked rows
constexpr int kOutC   = 512;         // stacked columns (7*73 + 1)
constexpr int kDModel = 1536;        // 3 * 512
constexpr int kMBlk   = 128;         // rows per workgroup = 8 waves * 16
// -2*ln(10000)/1536  (sinusoid frequency decay per column-pair index)
constexpr float kDivC   = -0.011992631f;
// exp(256 * kDivC) = 10000^(-1/3): per-jrow frequency decay factor
constexpr float kDivRow = 0.046415888f;
constexpr float kHalfPi = 1.57079632679f;   // cos(x) == sin(x + pi/2)
// Combined temporal table: T[p*7+q][j] = f(p,j) + f(q,j),  f = sin/cos(p*div)
constexpr int kTabRows  = 49;        // x_mark values are in [0,7)
constexpr int kTabElems = kTabRows * kDModel;
}

// seqs[c,h] = x[min(h/3 + h%3, SEQ-1), c], zero outside [0, kRows).
// Branch-free: clamped load + select (no saveexec around the global_load).
__device__ __forceinline__ float load_seq(const float* __restrict__ x, int c, int h) {
    int hc = h < 0 ? 0 : (h >= kRows ? kRows - 1 : h);
    int s  = hc / 3 + hc % 3;
    s = s > (kSeq - 1) ? (kSeq - 1) : s;
    float v = x[s * kCin + c];
    return ((unsigned)h < (unsigned)kRows) ? v : 0.0f;
}

// W[o,k] = kernels[o, k, 1]. Index clamp only (no zero-mask needed: WMMA column
// o only feeds output column o, and o >= 74 lanes are never stored).
__device__ __forceinline__ float load_w(const float* __restrict__ kern, int o, int kk) {
    int oc = o < kNumK ? o : kNumK - 1;
    return kern[oc * 24 + kk * 3 + 1];
}

// ---- Setup kernel: build combined temporal table into workspace (301 KB).
__global__ __launch_bounds__(256) void
build_temporal_table(float* __restrict__ T)
{
    const int i = blockIdx.x * 256 + threadIdx.x;
    if (i >= kTabElems) return;
    const int a = i / kDModel;          // pair index p*7+q
    const int j = i - a * kDModel;      // d_model column
    const int p = a / 7;
    const int q = a - p * 7;
    const float div = __expf((float)(j >> 1) * kDivC);
    const float ph  = (j & 1) ? kHalfPi : 0.0f;     // branch-free sin/cos
    T[i] = __sinf(fmaf((float)p, div, ph)) + __sinf(fmaf((float)q, div, ph));
}

// ---- Main fused kernel. USE_TAB: temporal via 2 coalesced table loads
// (1 TRANS/element total); else 5 TRANS/element inline (workspace fallback).
template <bool USE_TAB>
__global__ __launch_bounds__(256) void
DataEmbedding_26886495273553_kernel(const float* __restrict__ x,
                                    const float* __restrict__ kernels,
                                    const int*   __restrict__ xmark,
                                    const float* __restrict__ T,
                                    float*       __restrict__ out)
{
    const int wave  = threadIdx.x >> 5;                 // 0..7
    const int lane  = threadIdx.x & 31;
    const int lhalf = lane >> 4;                        // 0 | 1
    const int lmod  = lane & 15;
    const int c     = blockIdx.y;                       // channel 0..6
    const int r0    = blockIdx.x * kMBlk + wave * 16;   // M-tile row base

    // ---- A operand: 16x4 fp32, ISA layout
    //   lanes 0-15 : M=lane,    VGPR0=K0, VGPR1=K1
    //   lanes 16-31: M=lane-16, VGPR0=K2, VGPR1=K3
    // A[M][K] = seqs[c, r0+M + K - 4]   (conv padding (4,3))
    const int kb = lhalf * 2;
    v2f aLo, aHi;                                       // K=0..3 / K=4..7 halves
    aLo.x = load_seq(x, c, r0 + lmod + kb - 4);
    aLo.y = load_seq(x, c, r0 + lmod + kb - 3);
    aHi.x = load_seq(x, c, r0 + lmod + kb + 0);
    aHi.y = load_seq(x, c, r0 + lmod + kb + 1);

    // ---- Epilogue-invariant per-lane state (shared by all 5 N-tiles).
    // D layout: VGPR v -> row rbase+v, rbase = r0 (+8 for upper lane half).
    const int rbase = r0 + lhalf * 8;
    float sf[8];        // (float)s
    int   jr[8];        // r % 3
    int   offA[8];      // table element offset (row base + jrow*512), pair (m3,m2)
    int   offB[8];      // pair (m1,m0)
    float xf[8][4];     // fallback path: marks as float
    #pragma unroll
    for (int v = 0; v < 8; ++v) {
        const int r = rbase + v;
        const int s = r / 3;
        jr[v] = r - 3 * s;
        sf[v] = (float)s;
        const int4 xm = ((const int4* __restrict__)xmark)[s];  // one b128 load
        if constexpr (USE_TAB) {
            const int m0 = min(xm.x, 6), m1 = min(xm.y, 6);
            const int m2 = min(xm.z, 6), m3 = min(xm.w, 6);
            offA[v] = (m3 * 7 + m2) * kDModel + jr[v] * kOutC;
            offB[v] = (m1 * 7 + m0) * kDModel + jr[v] * kOutC;
        } else {
            xf[v][0] = (float)xm.w;  xf[v][1] = (float)xm.z;
            xf[v][2] = (float)xm.y;  xf[v][3] = (float)xm.x;
        }
    }

    // ---- 5 N-tiles cover o = 0..79 (o >= 74 masked at store)
    #pragma unroll
    for (int nt = 0; nt < 5; ++nt) {
        const int o = nt * 16 + lmod;

        // B operand 4x16: row-striped like C/D (VGPR0: K0 | K2, VGPR1: K1 | K3)
        v2f bLo, bHi;
        bLo.x = load_w(kernels, o, kb + 0);
        bLo.y = load_w(kernels, o, kb + 1);
        bHi.x = load_w(kernels, o, kb + 4);
        bHi.y = load_w(kernels, o, kb + 5);

        v8f acc = {};
        acc = __builtin_amdgcn_wmma_f32_16x16x4_f32(false, aLo, false, bLo,
                                                    (short)0, acc, false, false);
        acc = __builtin_amdgcn_wmma_f32_16x16x4_f32(false, aHi, false, bHi,
                                                    (short)0, acc, false, false);

        // GEMM column -> stacked output column
        const bool isMain  = (o < kNumK - 1);
        const bool isExtra = (o == kNumK - 1) & (c == 0);
        const int  col     = isMain ? (c * (kNumK - 1) + o) : (kOutC - 1);

        if (isMain | isExtra) {
            // j = jrow*512 + col:  j&1 == col&1 (lane-uniform),
            // j>>1 == jrow*256 + (col>>1)  =>  div(jrow) = div0 * kDivRow^jrow
            const float div0 = __expf((float)(col >> 1) * kDivC);
            const float div1 = div0 * kDivRow;
            const float div2 = div1 * kDivRow;
            // cos(x) == sin(x + pi/2): one cndmask per tile, no divergent
            // sin/cos branch per element.
            const float ph = (col & 1) ? kHalfPi : 0.0f;
            float* __restrict__ op = out + col;

            #pragma unroll
            for (int v = 0; v < 8; ++v) {
                const float dv = jr[v] == 0 ? div0 : (jr[v] == 1 ? div1 : div2);
                float add;
                if constexpr (USE_TAB) {
                    // pe (1 TRANS) + 2 coalesced table reads (lanes share the
                    // row; col is consecutive across the 16-lane half)
                    add = __sinf(fmaf(sf[v], dv, ph))
                        + T[offA[v] + col] + T[offB[v] + col];
                } else {
                    add = __sinf(fmaf(sf[v],    dv, ph))
                        + __sinf(fmaf(xf[v][0], dv, ph))
                        + __sinf(fmaf(xf[v][1], dv, ph))
                        + __sinf(fmaf(xf[v][2], dv, ph))
                        + __sinf(fmaf(xf[v][3], dv, ph));
                }
                op[(rbase + v) * kOutC] = acc[v] + add;
            }
        }
    }
}

extern "C" void kernel_launch(void* const* d_in, const int* in_sizes, int n_in,
                              void* d_out, int out_size, void* d_ws, size_t ws_size,
                              hipStream_t stream) {
    (void)in_sizes; (void)n_in; (void)out_size;
    const float* x       = (const float*)d_in[0];   // (1,4096,7,1) f32
    const float* kernels = (const float*)d_in[1];   // (74,8,3) f32
    const int*   xmark   = (const int*)d_in[2];     // (1,4096,4) i32
    float*       out     = (float*)d_out;           // (1,4096,1536) f32

    const dim3 grid(kRows / kMBlk, kCin);           // 96 x 7 workgroups
    constexpr size_t tabBytes = (size_t)kTabElems * sizeof(float);

    if (d_ws != nullptr && ws_size >= tabBytes) {
        float* T = (float*)d_ws;
        build_temporal_table<<<(kTabElems + 255) / 256, 256, 0, stream>>>(T);
        DataEmbedding_26886495273553_kernel<true>
            <<<grid, 256, 0, stream>>>(x, kernels, xmark, T, out);
    } else {
        DataEmbedding_26886495273553_kernel<false>
            <<<grid, 256, 0, stream>>>(x, kernels, xmark, nullptr, out);
    }
}